// LightGCNEncoder_14748917694875
// MI455X (gfx1250) — compile-verified
//
#include <hip/hip_runtime.h>
#include <stdint.h>

// LightGCN propagation on MI455X (gfx1250).
// Memory/latency-bound sparse gather (0.25 FLOP/byte): WMMA is inapplicable
// (random sparsity cannot share a WMMA B-tile across output rows without ~16x
// wasted work). CDNA5-specific optimization used instead:
//   * CSR-by-dst gather formulation -> zero float atomics
//   * X buffers (38.4MB each) are L2-resident (192MB L2) -> gather hits L2
//   * global_load_async_to_lds_b128 (ASYNCcnt) ring pipeline: D=6 pair-loads
//     (12 embedding rows) in flight per wave; async loads complete IN ORDER,
//     so "s_wait_asynccnt D" with a constant immediate guarantees the oldest
//     pair has landed in LDS -> gather latency overlapped with FMA compute.
//   * wave32-native layout: 1 wave per dst row, each lane owns a float2 of
//     the 64-dim row (32 lanes x 8B = one row per ds_load_b64).

#define DIM 64
#define SPMM_WAVES 4                  // waves per block
#define SPMM_THREADS (SPMM_WAVES * 32)
#define D_DEPTH 6                     // async pair-loads kept in flight
#define SLOTS 8                       // ring slots (pairs); pow2, > D_DEPTH

// ---------------- init: X0 = concat(user_emb, item_emb); acc(d_out) = X0 ----
__global__ void k_init(const float* __restrict__ uemb, const float* __restrict__ iemb,
                       float* __restrict__ X, float* __restrict__ acc,
                       long long nUser64, long long total) {
  long long i = (long long)blockIdx.x * blockDim.x + threadIdx.x;
  if (i >= total) return;
  float v = (i < nUser64) ? uemb[i] : iemb[i - nUser64];
  X[i] = v;
  acc[i] = v;
}

__global__ void k_zero(int* __restrict__ p, int n) {
  int i = blockIdx.x * blockDim.x + threadIdx.x;
  if (i < n) p[i] = 0;
}

// ---------------- CSR build ------------------------------------------------
__global__ void k_count(const int* __restrict__ dst, int* __restrict__ counts, int E) {
  int e = blockIdx.x * blockDim.x + threadIdx.x;
  if (e < E) atomicAdd(&counts[dst[e]], 1);
}

// Single-block hierarchical exclusive scan over N (~150k) counts.
// counts[] is read as input and overwritten with the row offsets (cursor).
__global__ void k_scan(int* __restrict__ counts, int* __restrict__ row_ptr,
                       int N, int E) {
  __shared__ int sdata[1024];
  const int t = threadIdx.x;
  const int chunk = (N + 1023) >> 10;
  const int c0 = t * chunk;
  int s = 0;
  for (int i = 0; i < chunk; ++i) {
    int idx = c0 + i;
    if (idx < N) s += counts[idx];
  }
  sdata[t] = s;
  __syncthreads();
  for (int off = 1; off < 1024; off <<= 1) {
    int v = 0;
    if (t >= off) v = sdata[t - off];
    __syncthreads();
    sdata[t] += v;
    __syncthreads();
  }
  int run = (t == 0) ? 0 : sdata[t - 1];
  for (int i = 0; i < chunk; ++i) {
    int idx = c0 + i;
    if (idx < N) {
      int c = counts[idx];
      row_ptr[idx] = run;
      counts[idx] = run;   // counts buffer becomes the fill cursor
      run += c;
    }
  }
  if (t == 0) row_ptr[N] = E;
}

__global__ void k_fill(const int* __restrict__ src, const int* __restrict__ dst,
                       const float* __restrict__ norm,
                       int* __restrict__ cursor, int* __restrict__ col,
                       float* __restrict__ wgt, int E) {
  int e = blockIdx.x * blockDim.x + threadIdx.x;
  if (e >= E) return;
  int pos = atomicAdd(&cursor[dst[e]], 1);
  col[pos] = src[e];
  wgt[pos] = norm[e];
}

// ---------------- SpMM: Xout[r] = sum_k w[k] * Xin[col[k]]; acc += Xout -----
// One wave per destination row. Neighbor rows (256B) staged through an LDS
// ring with async load-to-LDS. Each async b128 op moves one PAIR of rows:
// lanes 0-15 -> even row, lanes 16-31 -> odd row, 16B/lane.
__global__ void k_spmm(const int* __restrict__ row_ptr, const int* __restrict__ col,
                       const float* __restrict__ wgt, const float* __restrict__ Xin,
                       float* __restrict__ Xout, float* __restrict__ accOut,
                       int N, int isFinal) {
  __shared__ float stage[SPMM_WAVES][SLOTS * 2 * DIM];   // 4KB per wave
  const int lane = threadIdx.x & 31;
  const int wv = threadIdx.x >> 5;
  const int row = blockIdx.x * SPMM_WAVES + wv;
  if (row >= N) return;

  const int begin = row_ptr[row];
  const int end = row_ptr[row + 1];
  float ax = 0.f, ay = 0.f;

  // Low 32 bits of a flat shared-memory address are the LDS byte address.
  const uint32_t lds_base = (uint32_t)(uintptr_t)(&stage[wv][0]);
  const int sub = lane >> 4;                             // row within pair
  const uint32_t byteoff = (uint32_t)((lane & 15) << 4); // 16B per lane
  const float* sbase = &stage[wv][0];

  // Super-chunks of 32 neighbors (col/wgt held in per-lane registers).
  for (int base = begin; base < end; base += 32) {
    int m = end - base;
    if (m > 32) m = 32;
    int cc = 0; float ww = 0.f;
    if (lane < m) { cc = col[base + lane]; ww = wgt[base + lane]; }
    const int P = (m + 1) >> 1;                          // pairs in chunk

    auto issue = [&](int p) {                            // stage pair p
      int j = (p << 1) | sub;
      int cj = __shfl(cc, j, 32);                        // j>=m -> cc==0, benign
      uint64_t ga = (uint64_t)(uintptr_t)((const char*)(Xin + (size_t)cj * DIM) + byteoff);
      uint32_t la = lds_base + (uint32_t)((p & (SLOTS - 1)) << 9)
                             + (uint32_t)(sub << 8) + byteoff;
      asm volatile("global_load_async_to_lds_b128 %0, %1, off"
                   :: "v"(la), "v"(ga) : "memory");
    };
    auto consume = [&](int p) {                          // FMA pair p from LDS
      const float* sf = sbase + ((p & (SLOTS - 1)) << 7);
      int j0 = p << 1;
      float wj = __shfl(ww, j0, 32);
      float2 xv = *(const float2*)(sf + 2 * lane);
      ax = fmaf(wj, xv.x, ax); ay = fmaf(wj, xv.y, ay);
      if (j0 + 1 < m) {
        float wj1 = __shfl(ww, j0 + 1, 32);
        float2 xv1 = *(const float2*)(sf + DIM + 2 * lane);
        ax = fmaf(wj1, xv1.x, ax); ay = fmaf(wj1, xv1.y, ay);
      }
    };

    // Prologue: fill the pipeline.
    const int pre = (P < D_DEPTH) ? P : D_DEPTH;
    for (int p = 0; p < pre; ++p) issue(p);
    // Steady state: issue p+D, wait asynccnt<=D (in-order completion =>
    // pair p has landed), consume p. Constant immediate wait.
    int p = 0;
    for (; p + D_DEPTH < P; ++p) {
      issue(p + D_DEPTH);
      asm volatile("s_wait_asynccnt %0" :: "i"(D_DEPTH) : "memory");
      consume(p);
    }
    // Drain.
    asm volatile("s_wait_asynccnt 0x0" ::: "memory");
    for (; p < P; ++p) consume(p);
  }

  const size_t ofs = (size_t)row * DIM + 2 * lane;
  float2* ap = (float2*)(accOut + ofs);
  float2 a = *ap;
  a.x += ax; a.y += ay;
  if (isFinal) {            // fold the /(L+1) into the last layer; skip Xout
    a.x *= 0.25f; a.y *= 0.25f;
  } else {
    float2 res; res.x = ax; res.y = ay;
    *(float2*)(Xout + ofs) = res;
  }
  *ap = a;
}

// ---------------------------------------------------------------------------
extern "C" void kernel_launch(void* const* d_in, const int* in_sizes, int n_in,
                              void* d_out, int out_size, void* d_ws, size_t ws_size,
                              hipStream_t stream) {
  // setup_inputs order: users_all, items_all, edge_index[2,E], norm[E],
  //                     user_emb[nu,64], item_emb[ni,64]
  const int nu = in_sizes[0];
  const int ni = in_sizes[1];
  const int E  = in_sizes[3];               // edge count == len(norm)
  const int N  = nu + ni;

  const int*   eidx = (const int*)d_in[2];
  const int*   src  = eidx;                 // edge_index row 0
  const int*   dst  = eidx + E;             // edge_index row 1
  const float* norm = (const float*)d_in[3];
  const float* uemb = (const float*)d_in[4];
  const float* iemb = (const float*)d_in[5];
  float* out = (float*)d_out;               // acc, then final result

  // Workspace layout (~110 MB): Xa | Xb | row_ptr | cursor | col | wgt
  float* Xa      = (float*)d_ws;
  float* Xb      = Xa + (size_t)N * DIM;
  int*   row_ptr = (int*)(Xb + (size_t)N * DIM);
  int*   cursor  = row_ptr + (N + 1);
  int*   colA    = cursor + (N + 1);
  float* wgtA    = (float*)(colA + E);

  const long long total = (long long)N * DIM;
  const int tb = 256;

  k_init<<<(int)((total + tb - 1) / tb), tb, 0, stream>>>(
      uemb, iemb, Xa, out, (long long)nu * DIM, total);
  k_zero<<<(N + 1 + tb - 1) / tb, tb, 0, stream>>>(cursor, N + 1);
  k_count<<<(E + tb - 1) / tb, tb, 0, stream>>>(dst, cursor, E);
  k_scan<<<1, 1024, 0, stream>>>(cursor, row_ptr, N, E);
  k_fill<<<(E + tb - 1) / tb, tb, 0, stream>>>(src, dst, norm, cursor, colA, wgtA, E);

  const int spmm_blocks = (N + SPMM_WAVES - 1) / SPMM_WAVES;
  // Layer 1: Xa -> Xb ; Layer 2: Xb -> Xa ; Layer 3: final (no Xout, *0.25).
  k_spmm<<<spmm_blocks, SPMM_THREADS, 0, stream>>>(row_ptr, colA, wgtA, Xa, Xb, out, N, 0);
  k_spmm<<<spmm_blocks, SPMM_THREADS, 0, stream>>>(row_ptr, colA, wgtA, Xb, Xa, out, N, 0);
  k_spmm<<<spmm_blocks, SPMM_THREADS, 0, stream>>>(row_ptr, colA, wgtA, Xa, Xb, out, N, 1);
}